// NGCF_4337916969353
// MI455X (gfx1250) — compile-verified
//
#include <hip/hip_runtime.h>
#include <math.h>

// ---------------------------------------------------------------------------
// NGCF forward for MI455X (gfx1250, wave32, WMMA).
// Bandwidth-bound: ~173 MB HBM adjacency stream (~7.5us floor @23.3TB/s);
// ego/side (77 MB) stay L2-resident. GEMMs done in fp32 WMMA 16x16x4.
// ---------------------------------------------------------------------------

#define N_USER 100000
#define N_ITEM 50000
#define NTOT   150000            // N_USER + N_ITEM
#define NE     4800000
#define DD     64
#define NL     3
#define NB     4096

typedef __attribute__((ext_vector_type(2))) float v2f;
typedef __attribute__((ext_vector_type(8))) float v8f;

// ----------------------------- init / zero ---------------------------------

__global__ void ngcf_init_ego(const float* __restrict__ ue,
                              const float* __restrict__ ie,
                              float* __restrict__ ego) {
  int i = blockIdx.x * blockDim.x + threadIdx.x;   // NTOT*DD = 9.6M, fits int
  if (i >= NTOT * DD) return;
  ego[i] = (i < N_USER * DD) ? ue[i] : ie[i - N_USER * DD];
}

__global__ void ngcf_zero(float* __restrict__ p, int n4) {
  int i = blockIdx.x * blockDim.x + threadIdx.x;
  if (i >= n4) return;
  ((float4*)p)[i] = make_float4(0.f, 0.f, 0.f, 0.f);
}

// ------------------------------- SpMM --------------------------------------
// One wave32 per edge; each lane owns 2 consecutive floats of the 64-wide row.
// Atomic f32 adds resolve in L2 (side is L2-resident).

__global__ void ngcf_spmm(const float* __restrict__ val,
                          const int*   __restrict__ row,
                          const int*   __restrict__ col,
                          const float* __restrict__ ego,
                          float*       __restrict__ side) {
  unsigned tid  = blockIdx.x * blockDim.x + threadIdx.x;
  unsigned e    = tid >> 5;
  if (e >= NE) return;
  unsigned lane = tid & 31;

  // stream-ahead hint for the index arrays (global_prefetch_b8)
  if (e + 2048 < NE) {
    __builtin_prefetch(col + e + 2048, 0, 0);
    __builtin_prefetch(row + e + 2048, 0, 0);
  }

  float v = val[e];
  int   r = row[e];
  int   c = col[e];

  const float2 x = *(const float2*)(ego + (size_t)c * DD + lane * 2);
  float* dst = side + (size_t)r * DD + lane * 2;
  unsafeAtomicAdd(dst + 0, v * x.x);
  unsafeAtomicAdd(dst + 1, v * x.y);
}

// ------------------------ fused dual-GEMM + activation ----------------------
// ego_new = leakyrelu( [side | ego*side] @ [Wgc ; Wbi] + (bgc+bbi) , 0.2 )
// 16 nodes per block, 4 waves; wave w owns output columns [16w,16w+16).
// K = 128 consumed in 32 steps of V_WMMA_F32_16X16X4_F32.
//
// f32 WMMA lane layouts (ISA 7.12.2):
//   A 16x4 : lanes0-15 M=0..15 {v0:K=0, v1:K=1}; lanes16-31 {v0:K=2, v1:K=3}
//   B 4x16 : lanes0-15 N=0..15 {v0:K=0, v1:K=1}; lanes16-31 {v0:K=2, v1:K=3}
//   C 16x16: VGPR j -> lanes0-15 (M=j, N=lane), lanes16-31 (M=j+8, N=lane-16)

__launch_bounds__(128)
__global__ void ngcf_combine(const float* __restrict__ ego_in,
                             const float* __restrict__ side,
                             const float* __restrict__ Wgc,
                             const float* __restrict__ bgc,
                             const float* __restrict__ Wbi,
                             const float* __restrict__ bbi,
                             float* __restrict__ ego_out) {
  __shared__ float sA[16 * 132];      // padded stride: avoid LDS bank conflicts
  __shared__ float sB[128 * 64];
  __shared__ float sBias[64];

  const int tid  = threadIdx.x;
  const int base = blockIdx.x * 16;   // NTOT/16 = 9375 blocks exactly

  // Stage stacked weights [Wgc ; Wbi] : B[K][n]
  for (int i = tid; i < 128 * 64; i += 128) {
    int K = i >> 6, n = i & 63;
    sB[i] = (K < 64) ? Wgc[K * 64 + n] : Wbi[(K - 64) * 64 + n];
  }
  // Stage A = [ side | ego .* side ] for this block's 16 rows
  for (int i = tid; i < 16 * 64; i += 128) {
    int m = i >> 6, d = i & 63;
    float s = side[(base + m) * DD + d];
    float e = ego_in[(base + m) * DD + d];
    sA[m * 132 + d]      = s;
    sA[m * 132 + 64 + d] = s * e;
  }
  if (tid < 64) sBias[tid] = bgc[tid] + bbi[tid];
  __syncthreads();

  const int w       = tid >> 5;
  const int lane    = tid & 31;
  const int halfSel = lane >> 4;          // 0: K+0/1 , 1: K+2/3
  const int mrow    = lane & 15;          // A row (lanes) == B/C column offset
  const int koff    = halfSel * 2;
  const int n       = (w << 4) | mrow;    // output column 0..63

  v8f acc = {};
  #pragma unroll 8
  for (int kk = 0; kk < 32; ++kk) {
    const int Kb = kk * 4 + koff;
    v2f a, b;
    a.x = sA[mrow * 132 + Kb];
    a.y = sA[mrow * 132 + Kb + 1];
    b.x = sB[(Kb + 0) * 64 + n];
    b.y = sB[(Kb + 1) * 64 + n];
    acc = __builtin_amdgcn_wmma_f32_16x16x4_f32(
        /*neg_a=*/false, a, /*neg_b=*/false, b,
        /*c_mod=*/(short)0, acc, /*reuse_a=*/false, /*reuse_b=*/false);
  }

  const float bias = sBias[n];
  #pragma unroll
  for (int j = 0; j < 8; ++j) {
    float v = acc[j] + bias;
    v = fmaxf(v, 0.2f * v);               // leaky_relu(0.2)
    const int M = halfSel ? (j + 8) : j;
    ego_out[(size_t)(base + M) * DD + n] = v;   // row-local: in-place safe
  }
}

// ---------------------- gather (+ optional L2 normalize) --------------------
// One wave per output row; rows 0..4095 -> users, 4096..8191 -> items.
// Writes 64-column slice [colOff, colOff+64) of out[idx, 0:256].

__global__ void ngcf_gather(const float* __restrict__ ego,
                            const int* __restrict__ users,
                            const int* __restrict__ items,
                            float* __restrict__ out,
                            int colOff, int doNorm) {
  const int idx  = blockIdx.x * (blockDim.x >> 5) + (threadIdx.x >> 5);
  const int lane = threadIdx.x & 31;
  if (idx >= 2 * NB) return;
  const int r = (idx < NB) ? users[idx] : (N_USER + items[idx - NB]);

  const float2 x = *(const float2*)(ego + (size_t)r * DD + lane * 2);
  float s = 1.0f;
  if (doNorm) {
    float ss = x.x * x.x + x.y * x.y;
    #pragma unroll
    for (int off = 16; off > 0; off >>= 1) ss += __shfl_xor(ss, off, 32);
    s = 1.0f / fmaxf(sqrtf(ss), 1e-12f);
  }
  float* dst = out + (size_t)idx * (4 * DD) + colOff + lane * 2;
  dst[0] = x.x * s;
  dst[1] = x.y * s;
}

// ------------------------------- launcher ----------------------------------

extern "C" void kernel_launch(void* const* d_in, const int* in_sizes, int n_in,
                              void* d_out, int out_size, void* d_ws, size_t ws_size,
                              hipStream_t stream) {
  (void)in_sizes; (void)n_in; (void)out_size; (void)ws_size;

  const float* user_emb = (const float*)d_in[0];
  const float* item_emb = (const float*)d_in[1];
  const float* W_gc     = (const float*)d_in[2];   // [3,64,64]
  const float* b_gc     = (const float*)d_in[3];   // [3,64]
  const float* W_bi     = (const float*)d_in[4];
  const float* b_bi     = (const float*)d_in[5];
  const float* adj_val  = (const float*)d_in[6];
  const int*   adj_row  = (const int*)d_in[7];
  const int*   adj_col  = (const int*)d_in[8];
  const int*   users    = (const int*)d_in[9];
  const int*   items    = (const int*)d_in[10];
  float*       out      = (float*)d_out;

  float* ego  = (float*)d_ws;                         // [NTOT,64]  38.4 MB
  float* side = ego + (size_t)NTOT * DD;              // [NTOT,64]  38.4 MB

  const int nElem = NTOT * DD;

  ngcf_init_ego<<<(nElem + 255) / 256, 256, 0, stream>>>(user_emb, item_emb, ego);
  // layer-0 entry of all_embeddings is the raw (un-normalized) ego
  ngcf_gather<<<(2 * NB) / 8, 256, 0, stream>>>(ego, users, items, out, 0, 0);

  for (int k = 0; k < NL; ++k) {
    ngcf_zero<<<(nElem / 4 + 255) / 256, 256, 0, stream>>>(side, nElem / 4);
    ngcf_spmm<<<(NE * 32) / 256, 256, 0, stream>>>(adj_val, adj_row, adj_col, ego, side);
    ngcf_combine<<<NTOT / 16, 128, 0, stream>>>(
        ego, side,
        W_gc + (size_t)k * DD * DD, b_gc + (size_t)k * DD,
        W_bi + (size_t)k * DD * DD, b_bi + (size_t)k * DD,
        ego);
    ngcf_gather<<<(2 * NB) / 8, 256, 0, stream>>>(ego, users, items, out,
                                                  DD * (k + 1), 1);
  }
}